// RCModel_52003464020486
// MI455X (gfx1250) — compile-verified
//
#include <hip/hip_runtime.h>

typedef __attribute__((ext_vector_type(2))) float v2f;
typedef __attribute__((ext_vector_type(4))) float v4f;
typedef __attribute__((ext_vector_type(8))) float v8f;

#define NSTATE 20
#define NAUG   24    // augmented K: 20 state + [T, 1, 0, 0]
#define NROOMS 18
#define LCH    128   // steps per chunk (power of two: 7 squarings -> P^128)
#define NCOLS  16    // chunks batched per wave (WMMA N dimension)
#define XSTRIDE 30   // cols 0..23 = state, 24..29 = dummy sink for hl==1 lanes

static __device__ __forceinline__ v8f wmma4(v2f a, v2f b, v8f c) {
  // V_WMMA_F32_16X16X4_F32: D(16x16,f32) = A(16x4,f32) * B(4x16,f32) + C
  return __builtin_amdgcn_wmma_f32_16x16x4_f32(
      false, a, false, b, (short)0, c, false, false);
}

// ---------------------------------------------------------------------------
// Setup: augmented matrix [A | b0 | c | 0 | 0] (32x24, zero padded rows),
// P = homogeneous RK step matrix, P^128 via 7 squarings (final in W0).
// ---------------------------------------------------------------------------
__global__ __launch_bounds__(64) void setup_kernel(
    const float* __restrict__ A, const float* __restrict__ B,
    const float* __restrict__ iv, const float* __restrict__ cool,
    const float* __restrict__ gain, const int* __restrict__ action,
    float* apad, float* P, float* W0, float* W1, float* out)
{
  __shared__ float Qs[NROOMS];
  const int tid = threadIdx.x;  // blockDim.x == 64
  const float af = (float)action[0];
  if (tid < NROOMS) Qs[tid] = 10.0f * (-cool[tid] * af + gain[tid]);
  if (tid < 32) {
    for (int j = 0; j < NSTATE; ++j)
      apad[tid * NAUG + j] = (tid < NSTATE) ? A[tid * NSTATE + j] : 0.0f;
    apad[tid * NAUG + 20] = (tid < NSTATE) ? B[tid * (1 + NROOMS)] : 0.0f; // b0
    apad[tid * NAUG + 22] = 0.0f;
    apad[tid * NAUG + 23] = 0.0f;
  }
  __syncthreads();
  if (tid < 32) {
    float s = 0.0f;
    if (tid < NSTATE)
      for (int j = 0; j < NROOMS; ++j) s += B[tid * (1 + NROOMS) + 1 + j] * Qs[j];
    apad[tid * NAUG + 21] = s;                                            // c
  }
  if (tid < NSTATE) out[tid] = iv[tid];  // output row 0 = iv
  __syncthreads();

  // Column tid of P: push basis vector e_tid through one homogeneous RK step.
  if (tid < NSTATE) {
    float x[NSTATE], k1[NSTATE], k2[NSTATE], k3[NSTATE], k4[NSTATE], tm[NSTATE];
    for (int i = 0; i < NSTATE; ++i) x[i] = (i == tid) ? 1.0f : 0.0f;
    for (int i = 0; i < NSTATE; ++i) { float s = 0; for (int j = 0; j < NSTATE; ++j) s += apad[i*NAUG+j] * x[j];  k1[i] = s; }
    for (int i = 0; i < NSTATE; ++i) tm[i] = x[i] + 10.0f * k1[i];
    for (int i = 0; i < NSTATE; ++i) { float s = 0; for (int j = 0; j < NSTATE; ++j) s += apad[i*NAUG+j] * tm[j]; k2[i] = s; }
    for (int i = 0; i < NSTATE; ++i) tm[i] = x[i] + 30.0f * k2[i] - 10.0f * k1[i];
    for (int i = 0; i < NSTATE; ++i) { float s = 0; for (int j = 0; j < NSTATE; ++j) s += apad[i*NAUG+j] * tm[j]; k3[i] = s; }
    for (int i = 0; i < NSTATE; ++i) tm[i] = x[i] + 30.0f * (k1[i] - k2[i] + k3[i]);
    for (int i = 0; i < NSTATE; ++i) { float s = 0; for (int j = 0; j < NSTATE; ++j) s += apad[i*NAUG+j] * tm[j]; k4[i] = s; }
    for (int i = 0; i < NSTATE; ++i)
      P[i * NSTATE + tid] = x[i] + 3.75f * (k1[i] + 3.0f * (k2[i] + k3[i]) + k4[i]);
  }
  __syncthreads();

  for (int m = 0; m < 7; ++m) {   // P^128, final lands in W0
    const float* src = (m == 0) ? P : ((m & 1) ? W0 : W1);
    float* dst = (m & 1) ? W1 : W0;
    for (int e = tid; e < NSTATE * NSTATE; e += blockDim.x) {
      int i = e / NSTATE, j = e % NSTATE;
      float s = 0.0f;
      for (int k = 0; k < NSTATE; ++k) s += src[i * NSTATE + k] * src[k * NSTATE + j];
      dst[e] = s;
    }
    __syncthreads();
  }
}

// ---------------------------------------------------------------------------
// Pass 2: sequential combine  X_{c+1} = P^L * X_c + G_c   (single wave).
// ---------------------------------------------------------------------------
__global__ __launch_bounds__(32) void scan_kernel(
    const float* __restrict__ iv, const float* __restrict__ PL,
    const float* __restrict__ G, float* __restrict__ Xc, int C)
{
  const int l = threadIdx.x;
  const int lr = (l < NSTATE) ? l : 0;
  float row[NSTATE];
#pragma unroll
  for (int j = 0; j < NSTATE; ++j) row[j] = PL[lr * NSTATE + j];
  float x = (l < NSTATE) ? iv[l] : 0.0f;
  for (int c = 0; c < C; ++c) {
    if (l < NSTATE) Xc[c * NSTATE + l] = x;
    if (c < C - 1) {
      float s = G[c * NSTATE + lr];
#pragma unroll
      for (int j = 0; j < NSTATE; ++j) s += row[j] * __shfl(x, j, 32);
      x = (l < NSTATE) ? s : 0.0f;
    }
  }
}

// ---------------------------------------------------------------------------
// WMMA chunk engine: 16 chunks/wave; augmented matvec (K=24) via
// 2 M-tiles x 6 K-slices of v_wmma_f32_16x16x4_f32; LDS tile chunk-major.
// All LDS stores are unconditional (hl==1 lanes write to dummy columns).
// ---------------------------------------------------------------------------
static __device__ __forceinline__ void mv24(const v2f (&af)[2][6], const float* xt,
                                            int ldbase, v8f& o0, v8f& o1)
{
  v8f z = {0, 0, 0, 0, 0, 0, 0, 0};
  o0 = z; o1 = z;
#pragma unroll
  for (int kb = 0; kb < 6; ++kb) {
    v2f b = *(const v2f*)&xt[ldbase + 4 * kb];            // ds_load_b64
    o0 = wmma4(af[0][kb], b, o0);     // output rows i = 0..15
    o1 = wmma4(af[1][kb], b, o1);     // output rows i = 16..19 (rest pad 0)
  }
}

static __device__ __forceinline__ void store_xt(float* xt, int sbase, int t1base,
                                                const v8f& a0, const v8f& a1)
{
#pragma unroll
  for (int v = 0; v < 8; v += 2) {
    v2f p = {a0[v], a0[v + 1]};
    *(v2f*)&xt[sbase + v] = p;                             // ds_store_b64
  }
  v2f p0 = {a1[0], a1[1]};
  v2f p1 = {a1[2], a1[3]};
  *(v2f*)&xt[t1base] = p0;         // hl==0: cols 16..19, hl==1: dummy 24..27
  *(v2f*)&xt[t1base + 2] = p1;
}

static __device__ __forceinline__ void store_T(float* xt, int tbase, float T)
{
  v2f p = {T, 1.0f};
  *(v2f*)&xt[tbase] = p;           // hl==0: cols 20,21 = [T,1]; hl==1: dummy 28,29
}

template <bool PASS3>
__global__ __launch_bounds__(256, 1) void chunk_kernel(
    const float* __restrict__ tout, const float* __restrict__ apad,
    const float* __restrict__ Xc, float* __restrict__ G,
    float* __restrict__ out, int C, int NT, int Ttot)
{
  __shared__ __align__(16) float xts[8][NCOLS * XSTRIDE];
  const int lane = threadIdx.x & 31;
  const int wid  = threadIdx.x >> 5;
  const int hl = lane >> 4, ml = lane & 15;
  const int c0 = (blockIdx.x * (blockDim.x >> 5) + wid) * NCOLS;
  if (c0 >= C) return;
  float* xt = xts[wid];

  // Per-lane LDS bases (float indices), computed once.
  const int rowb   = ml * XSTRIDE;
  const int ldbase = rowb + 2 * hl;              // B-frag load base
  const int sbase  = rowb + 8 * hl;              // slot store base (i = v + 8*hl)
  const int t1base = rowb + (hl ? 24 : 16);      // tile-1 store (or dummy)
  const int tbase  = rowb + (hl ? 28 : 20);      // [T,1] store (or dummy)

  // Constant augmented-A fragments (16x4 f32 A-operand layout) in registers.
  v2f af[2][6];
#pragma unroll
  for (int t = 0; t < 2; ++t)
#pragma unroll
    for (int kb = 0; kb < 6; ++kb) {
      const int rowi = 16 * t + ml;
      const int k0 = 4 * kb + 2 * hl;
      af[t][kb].x = apad[rowi * NAUG + k0];
      af[t][kb].y = apad[rowi * NAUG + k0 + 1];
    }

  const int myc = c0 + ml;
  const int ccl = (myc < C) ? myc : (C - 1);
  const bool rowActive = (myc < C);
  const int base = ccl * LCH;
  int Lr = Ttot - base; if (Lr > LCH) Lr = LCH;

  // zero augmented rows 22,23 once (hl==1 lanes hit dummy cols 24,25)
  { v2f z2 = {0.0f, 0.0f}; *(v2f*)&xt[rowb + (hl ? 24 : 22)] = z2; }

  v8f x0 = {0, 0, 0, 0, 0, 0, 0, 0};
  v8f x1 = {0, 0, 0, 0, 0, 0, 0, 0};
  if (PASS3) {
    const v4f* xc4 = (const v4f*)(Xc + ccl * NSTATE);   // 16B aligned (20*c % 4 == 0)
    v4f qa = xc4[2 * hl], qb = xc4[2 * hl + 1];
#pragma unroll
    for (int v = 0; v < 4; ++v) { x0[v] = qa[v]; x0[4 + v] = qb[v]; }
    if (hl == 0) {
      v4f qc = xc4[4];
#pragma unroll
      for (int v = 0; v < 4; ++v) x1[v] = qc[v];
    }
  }
  store_xt(xt, sbase, t1base, x0, x1);

  float a = tout[base];
  for (int s = 0; s < LCH; ++s) {
    int bi = base + s + 1; if (bi > NT - 1) bi = NT - 1;
    if ((s & 15) == 0) {
      int pf = base + s + 48; if (pf > NT - 1) pf = NT - 1;
      __builtin_prefetch(tout + pf, 0, 1);    // global_prefetch_b8
    }
    const float b  = tout[bi];
    const float dT = b - a;

    v8f k0a, k0b, acc0, acc1, D0, D1;

    // k1
    store_T(xt, tbase, a);
    mv24(af, xt, ldbase, k0a, k0b);
    acc0 = k0a; acc1 = k0b;
    { v8f t0, t1;
#pragma unroll
      for (int v = 0; v < 8; ++v) { t0[v] = x0[v] + 10.0f * k0a[v]; t1[v] = x1[v] + 10.0f * k0b[v]; }
      store_xt(xt, sbase, t1base, t0, t1); }
    D0 = k0a; D1 = k0b;                       // D = k1 (will become k1 - k2)

    // k2
    store_T(xt, tbase, a + dT * (1.0f / 3.0f));
    mv24(af, xt, ldbase, k0a, k0b);
    { v8f t0, t1;
#pragma unroll
      for (int v = 0; v < 8; ++v) {
        t0[v] = x0[v] + 30.0f * k0a[v] - 10.0f * D0[v];
        t1[v] = x1[v] + 30.0f * k0b[v] - 10.0f * D1[v];
        acc0[v] += 3.0f * k0a[v]; acc1[v] += 3.0f * k0b[v];
        D0[v] -= k0a[v];          D1[v] -= k0b[v];        // D = k1 - k2
      }
      store_xt(xt, sbase, t1base, t0, t1); }

    // k3
    store_T(xt, tbase, a + dT * (2.0f / 3.0f));
    mv24(af, xt, ldbase, k0a, k0b);
    { v8f t0, t1;
#pragma unroll
      for (int v = 0; v < 8; ++v) {
        t0[v] = x0[v] + 30.0f * (D0[v] + k0a[v]);
        t1[v] = x1[v] + 30.0f * (D1[v] + k0b[v]);
        acc0[v] += 3.0f * k0a[v]; acc1[v] += 3.0f * k0b[v];
      }
      store_xt(xt, sbase, t1base, t0, t1); }

    // k4 + state update
    store_T(xt, tbase, b);
    mv24(af, xt, ldbase, k0a, k0b);
#pragma unroll
    for (int v = 0; v < 8; ++v) {
      x0[v] = x0[v] + 3.75f * (acc0[v] + k0a[v]);
      x1[v] = x1[v] + 3.75f * (acc1[v] + k0b[v]);
    }
    store_xt(xt, sbase, t1base, x0, x1);

    if (PASS3) {
      if (rowActive && s < Lr) {
        float* rp = out + (base + s + 1) * NSTATE;      // row base, 16B aligned
        v4f* rp4 = (v4f*)(rp + 8 * hl);
        v4f qa, qb;
#pragma unroll
        for (int v = 0; v < 4; ++v) { qa[v] = x0[v]; qb[v] = x0[4 + v]; }
        rp4[0] = qa; rp4[1] = qb;                        // global_store_b128
        if (hl == 0) {
          v4f qc;
#pragma unroll
          for (int v = 0; v < 4; ++v) qc[v] = x1[v];
          *(v4f*)(rp + 16) = qc;
        }
      }
    }
    a = b;
  }

  if (!PASS3) {
    if (rowActive) {
      float* gp = G + myc * NSTATE;
      v4f* gp4 = (v4f*)(gp + 8 * hl);
      v4f qa, qb;
#pragma unroll
      for (int v = 0; v < 4; ++v) { qa[v] = x0[v]; qb[v] = x0[4 + v]; }
      gp4[0] = qa; gp4[1] = qb;
      if (hl == 0) {
        v4f qc;
#pragma unroll
        for (int v = 0; v < 4; ++v) qc[v] = x1[v];
        *(v4f*)(gp + 16) = qc;
      }
    }
  }
}

// ---------------------------------------------------------------------------
extern "C" void kernel_launch(void* const* d_in, const int* in_sizes, int n_in,
                              void* d_out, int out_size, void* d_ws, size_t ws_size,
                              hipStream_t stream)
{
  const float* A    = (const float*)d_in[0];
  const float* B    = (const float*)d_in[1];
  const float* iv   = (const float*)d_in[2];
  const float* tout = (const float*)d_in[3];
  const float* cool = (const float*)d_in[4];
  const float* gain = (const float*)d_in[5];
  const int*   act  = (const int*)d_in[8];
  float* out = (float*)d_out;
  float* ws  = (float*)d_ws;

  const int NT   = in_sizes[3];          // 500000
  const int Ttot = NT - 1;               // RK steps
  const int C    = (Ttot + LCH - 1) / LCH;

  float* apad = ws + 0;                  // 32 x 24 augmented [A | b0 | c | 0 | 0]
  float* P    = ws + 768;                // 20x20
  float* W0   = ws + 1168;               // 20x20 (final P^128)
  float* W1   = ws + 1568;               // 20x20
  float* G    = ws + 2048;               // C x 20
  float* Xc   = G + (size_t)C * NSTATE;  // C x 20

  setup_kernel<<<1, 64, 0, stream>>>(A, B, iv, cool, gain, act, apad, P, W0, W1, out);

  const int chunksPerBlock = 8 * NCOLS;  // 8 waves x 16 chunks
  const int grid = (C + chunksPerBlock - 1) / chunksPerBlock;

  chunk_kernel<false><<<grid, 256, 0, stream>>>(tout, apad, Xc, G, out, C, NT, Ttot);
  scan_kernel<<<1, 32, 0, stream>>>(iv, W0, G, Xc, C);
  chunk_kernel<true><<<grid, 256, 0, stream>>>(tout, apad, Xc, G, out, C, NT, Ttot);
}